// All2AllDenseEmbedding_76828374991711
// MI455X (gfx1250) — compile-verified
//
#include <hip/hip_runtime.h>

// Embedding gather: out[r, :] = table[idx[r], :]
//   rows  = 4096*26*8 = 851,968
//   EMB   = 64 floats = 256 B per row = 16 lanes x b128
// Pure bandwidth problem (~0.44 GB total -> ~19us at 23.3 TB/s).
// Table reads: regular temporal (mostly resident in 192MB L2).
// Output: streamed once -> non-temporal stores (don't evict the table).

#define TOTAL_ROWS 851968LL
#define EMB 64

typedef float v4f __attribute__((ext_vector_type(4)));

// Pointee type per hipcc diagnostic: 'int __attribute__((vector_size(16)))'
typedef int v4i __attribute__((vector_size(4 * sizeof(int))));
typedef __attribute__((address_space(1))) v4i g_v4i;   // global
typedef __attribute__((address_space(3))) v4i l_v4i;   // LDS

// gfx12+ CPol: TH in [2:0]; TH=1 -> non-temporal.
#define CPOL_RT 0
#define CPOL_NT 1

#if __has_builtin(__builtin_amdgcn_global_load_async_to_lds_b128) && \
    __has_builtin(__builtin_amdgcn_global_store_async_from_lds_b128)
#define HAVE_ASYNC_LDS 1
#endif

#if __has_builtin(__builtin_amdgcn_s_wait_asynccnt)
#define WAIT_ASYNCCNT0() __builtin_amdgcn_s_wait_asynccnt(0)
#else
#define WAIT_ASYNCCNT0() asm volatile("s_wait_asynccnt 0x0" ::: "memory")
#endif

#ifdef HAVE_ASYNC_LDS
// ---------------------------------------------------------------------------
// CDNA5 async data mover path. 256 threads = 8 waves. Each wave DMAs 16 rows
// (8 x GLOBAL_LOAD_ASYNC_TO_LDS_B128, 512 B each) into its 4 KB LDS stage —
// no VGPR landing registers — waits ASYNCcnt==0, then streams the contiguous
// 4 KB out with 8 x GLOBAL_STORE_ASYNC_FROM_LDS_B128 (NT, one base address +
// literal INST_OFFSET stepping both LDS and global sides by 512 B).
// 128 rows per block; 6656 blocks.
// ---------------------------------------------------------------------------
__global__ __launch_bounds__(256) void gather_async_kernel(
    const long long* __restrict__ idx,
    const float* __restrict__ table,
    float* __restrict__ out)
{
    __shared__ float stage[8 * 1024];          // 8 waves * 4 KB
    const int tid    = threadIdx.x;
    const int wave   = tid >> 5;
    const int lane   = tid & 31;
    const int lane16 = lane & 15;              // 16 B chunk within a row
    const int half   = lane >> 4;              // which of 2 rows per instr
    const long long rbase = (long long)blockIdx.x * 128 + (long long)wave * 16;

    float* wstage = &stage[wave * 1024];

    // Gather: random global -> LDS (per-j addresses; rows are random).
#pragma unroll
    for (int j = 0; j < 8; ++j) {
        const long long r = rbase + j * 2 + half;
        const long long t = idx[r];
        const float* gp = table + t * EMB + lane16 * 4;   // 16 B per lane
        float*       lp = wstage + j * 128 + lane * 4;
        __builtin_amdgcn_global_load_async_to_lds_b128(
            (g_v4i*)gp, (l_v4i*)lp, /*offset=*/0, CPOL_RT);
    }

    WAIT_ASYNCCNT0();   // gathered rows now resident in LDS

    // Drain: contiguous LDS -> global. INST_OFFSET advances BOTH the LDS and
    // global addresses,
    // so one base address pair serves all 8 stores. Offsets must be literal
    // constants (frontend ICE check), hence the macro expansion.
    {
        float* gp0 = out + rbase * EMB + lane * 4;
        float* lp0 = wstage + lane * 4;
#define ASYNC_ST(OFF)                                                   \
        __builtin_amdgcn_global_store_async_from_lds_b128(              \
            (g_v4i*)gp0, (l_v4i*)lp0, (OFF), CPOL_NT)
        ASYNC_ST(0);
        ASYNC_ST(512);
        ASYNC_ST(1024);
        ASYNC_ST(1536);
        ASYNC_ST(2048);
        ASYNC_ST(2560);
        ASYNC_ST(3072);
        ASYNC_ST(3584);
#undef ASYNC_ST
    }
    // S_ENDPGM performs an implicit wait-idle: stores drain before wave exit.
}
#endif  // HAVE_ASYNC_LDS

// ---------------------------------------------------------------------------
// Fallback: direct VGPR gather (b128 loads, NT b128 stores), 4 independent
// chains per thread. 64 rows per 256-thread block.
// ---------------------------------------------------------------------------
__global__ __launch_bounds__(256) void gather_direct_kernel(
    const long long* __restrict__ idx,
    const float* __restrict__ table,
    float* __restrict__ out)
{
    const int tid    = threadIdx.x;
    const int group  = tid >> 4;
    const int lane16 = tid & 15;
    const long long rbase = (long long)blockIdx.x * 64 + group;

    long long r[4], t[4];
#pragma unroll
    for (int u = 0; u < 4; ++u) r[u] = rbase + u * 16;
#pragma unroll
    for (int u = 0; u < 4; ++u) t[u] = idx[r[u]];

    v4f v[4];
#pragma unroll
    for (int u = 0; u < 4; ++u)
        v[u] = *(const v4f*)(table + t[u] * EMB + lane16 * 4);

#pragma unroll
    for (int u = 0; u < 4; ++u)
        __builtin_nontemporal_store(v[u], (v4f*)(out + r[u] * EMB + lane16 * 4));
}

// ---------------------------------------------------------------------------
extern "C" void kernel_launch(void* const* d_in, const int* in_sizes, int n_in,
                              void* d_out, int out_size, void* d_ws, size_t ws_size,
                              hipStream_t stream) {
    (void)in_sizes; (void)n_in; (void)out_size; (void)d_ws; (void)ws_size;
    const long long* idx   = (const long long*)d_in[0];  // int64 indices [B,S,N]
    const float*     table = (const float*)d_in[1];      // f32 [VOCAB, 64]
    float*           out   = (float*)d_out;              // f32 [B,S,N,64]

#ifdef HAVE_ASYNC_LDS
    gather_async_kernel<<<(int)(TOTAL_ROWS / 128), 256, 0, stream>>>(idx, table, out);
#else
    gather_direct_kernel<<<(int)(TOTAL_ROWS / 64), 256, 0, stream>>>(idx, table, out);
#endif
}